// ROIAlign_86517821211843
// MI455X (gfx1250) — compile-verified
//
#include <hip/hip_runtime.h>
#include <hip/hip_bf16.h>

// ROI-Align 1D for MI455X (gfx1250, wave32).
//
// Shapes (fixed by the harness): input (16, 256, 2048) f32, rois (4096, 3) f32,
// feature_dim = 32, ratio = 2, output (4096, 256, 32) f32.
//
// Strategy: output-BW-bound kernel (134 MB writes vs 33.5 MB L2-resident input).
// Each wave computes one 16(bin) x 16(channel) tile of one roi. Per K=4 chunk it
// gathers+lerps 4 samples per channel into a WMMA B-fragment and multiplies by a
// 0.5-scaled selection matrix A so that V_WMMA_F32_16X16X4_F32 performs the
// mean-over-ratio reduction and the (sample,channel)->(bin,channel) transform,
// accumulating the 16x16 output tile in the C-fragment across 8 chunks.

typedef __attribute__((ext_vector_type(2))) float v2f;
typedef __attribute__((ext_vector_type(8))) float v8f;

#define ROI_CH   256
#define ROI_T    2048
#define ROI_FD   32   // feature_dim
// ratio = 2 baked into the sampling formula below.

__device__ __forceinline__ float roi_sample(const float* __restrict__ row,
                                            float start, float bin_size,
                                            int f, int r) {
    // t = start + (f + (r + 0.5)/ratio) * bin_size, ratio = 2
    float t = start + ((float)f + ((float)r + 0.5f) * 0.5f) * bin_size;
    bool valid = (t >= -1.0f) && (t <= (float)ROI_T);
    float tc = fminf(fmaxf(t, 0.0f), (float)(ROI_T - 1));
    int lo = (int)tc;                       // tc >= 0, trunc == floor
    lo = lo < (ROI_T - 1) ? lo : (ROI_T - 1);
    int hi = (lo + 1) < (ROI_T - 1) ? (lo + 1) : (ROI_T - 1);
    float frac = tc - (float)lo;
    float vlo = row[lo];
    float vhi = row[hi];
    float v = fmaf(vhi - vlo, frac, vlo);   // vlo*(1-frac) + vhi*frac
    return valid ? v : 0.0f;
}

__global__ __launch_bounds__(256) void
roialign_wmma_kernel(const float* __restrict__ input,   // (16, 256, 2048)
                     const float* __restrict__ rois,    // (N, 3)
                     float* __restrict__ out,           // (N, 256, 32)
                     int N) {
    const int n    = blockIdx.x;
    if (n >= N) return;
    const int wave = threadIdx.x >> 5;   // 0..7
    const int lane = threadIdx.x & 31;
    const int h    = lane >> 4;          // lane half: 0/1
    const int m    = lane & 15;          // 0..15

    const int   b        = (int)rois[3 * n + 0];
    const float start    = rois[3 * n + 1];
    const float end      = rois[3 * n + 2];
    const float roi_len  = fmaxf(end - start, 1.0f);
    const float bin_size = roi_len / (float)ROI_FD;

    const float* __restrict__ inB = input + (size_t)b * ROI_CH * ROI_T;

    // 32 tiles per roi: fhalf in {0,1} x ctile in {0..15}; 8 waves -> 4 tiles/wave.
    for (int tile = wave; tile < 32; tile += 8) {
        const int fhalf = tile >> 4;      // which 16-bin half
        const int ctile = tile & 15;      // which 16-channel group
        const int c     = ctile * 16 + m; // this lane's channel column
        const float* __restrict__ row = inB + (size_t)c * ROI_T;

        v8f acc = {};  // C fragment starts at 0, accumulates over 8 chunks

        #pragma unroll
        for (int kk = 0; kk < 8; ++kk) {
            // This lane's K rows are samples s = 4*kk + 2*h + {0,1};
            // both belong to local bin fl = s>>1 = 2*kk + h, with r = 0,1.
            const int fl = 2 * kk + h;            // local bin 0..15
            const int f  = fhalf * 16 + fl;       // global bin 0..31

            v2f bfrag;
            bfrag.x = roi_sample(row, start, bin_size, f, 0);
            bfrag.y = roi_sample(row, start, bin_size, f, 1);

            // A[m, 2h+v] = 0.5 iff output row m == (4kk+2h+v)>>1 == 2kk+h,
            // identical for v=0,1 -> mean over the 2 samples.
            const float aw = (m == fl) ? 0.5f : 0.0f;
            v2f afrag;
            afrag.x = aw;
            afrag.y = aw;

            acc = __builtin_amdgcn_wmma_f32_16x16x4_f32(
                /*neg_a=*/false, afrag, /*neg_b=*/false, bfrag,
                /*c_mod=*/(short)0, acc, /*reuse_a=*/false, /*reuse_b=*/false);
        }

        // C layout: VGPR v -> row M = v + 8h (bin), col N = m (channel).
        // out[n, c, f] with f contiguous; this lane owns 8 consecutive f values
        // starting at fhalf*16 + 8h -> two 16B stores.
        float* __restrict__ o =
            out + ((size_t)n * ROI_CH + (size_t)ctile * 16 + m) * ROI_FD
                + fhalf * 16 + 8 * h;
        float4 lo4 = make_float4(acc[0], acc[1], acc[2], acc[3]);
        float4 hi4 = make_float4(acc[4], acc[5], acc[6], acc[7]);
        *(float4*)(o + 0) = lo4;
        *(float4*)(o + 4) = hi4;
    }
}

extern "C" void kernel_launch(void* const* d_in, const int* in_sizes, int n_in,
                              void* d_out, int out_size, void* d_ws, size_t ws_size,
                              hipStream_t stream) {
    const float* input = (const float*)d_in[0];
    const float* rois  = (const float*)d_in[1];
    float* out         = (float*)d_out;
    const int N = in_sizes[1] / 3;   // 4096

    dim3 grid(N);
    dim3 block(256);
    roialign_wmma_kernel<<<grid, block, 0, stream>>>(input, rois, out, N);
}